// Lego_Message_Passing_Artificial_56238301774550
// MI455X (gfx1250) — compile-verified
//
#include <hip/hip_runtime.h>
#include <hip/hip_bf16.h>
#include <cstdint>
#include <cstddef>

typedef __attribute__((ext_vector_type(2))) float v2f;
typedef __attribute__((ext_vector_type(8))) float v8f;

#define NODE_DIM 64
#define EDGE_DIM 64
#define HIDDEN   128
#define BB       2
#define NN       512

// ---------------------------------------------------------------------------
// Kernel A: hi = node_feature @ W1 + b_edge ; hj = node_feature @ W2
// One 128-thread block per (b,n) row. Trivial FLOP count.
// ---------------------------------------------------------------------------
__global__ __launch_bounds__(128) void hij_kernel(
    const float* __restrict__ nf, const float* __restrict__ W_edge,
    const float* __restrict__ b_edge,
    float* __restrict__ hi, float* __restrict__ hj)
{
    __shared__ float row[NODE_DIM];
    const int r = blockIdx.x;           // 0 .. B*N-1
    const int h = threadIdx.x;          // 0 .. 127
    if (h < NODE_DIM) row[h] = nf[(size_t)r * NODE_DIM + h];
    __syncthreads();

    const float* W1 = W_edge;                        // rows 0..63
    const float* W2 = W_edge + NODE_DIM * HIDDEN;    // rows 64..127
    float a1 = b_edge[h];   // fold edge bias into hi
    float a2 = 0.f;
#pragma unroll 8
    for (int d = 0; d < NODE_DIM; ++d) {
        float v = row[d];
        a1 = fmaf(v, W1[d * HIDDEN + h], a1);
        a2 = fmaf(v, W2[d * HIDDEN + h], a2);
    }
    hi[(size_t)r * HIDDEN + h] = a1;
    hj[(size_t)r * HIDDEN + h] = a2;
}

// ---------------------------------------------------------------------------
// Kernel B: the big fused kernel.
//   he   = edge_feature[b,i,:,:] @ W3            (WMMA f32 16x16x4)
//   e    = relu(hi[b,i] + hj[b,j] + he + b_edge) -> edge_h (268 MB stream)
//   msg  = sum_j adjacency[b,i,j] * e            -> message[b,i,:]
// One 1024-thread block (32 waves) per (b,i); wave w owns j-tile [16w,16w+16).
// W3 staged in LDS pre-swizzled into WMMA B-fragments (one ds_load_b64 each).
// ---------------------------------------------------------------------------
__global__ __launch_bounds__(1024) void edge_kernel(
    const float* __restrict__ ef, const float* __restrict__ adj,
    const float* __restrict__ W_edge,
    const float* __restrict__ hi, const float* __restrict__ hj,
    float* __restrict__ edge_h, float* __restrict__ message)
{
    __shared__ float lds_w3[16 * 8 * 32 * 2];   // 32 KB: [t][n][lane] -> v2f
    __shared__ float lds_msg[HIDDEN];

    const int tid  = threadIdx.x;
    const int lane = tid & 31;
    const int wave = tid >> 5;          // 0..31
    const int hf   = lane >> 4;         // half-wave select (0/1)
    const int l16  = lane & 15;

    const int bi = blockIdx.x;          // flattened (b*N + i)

    // --- stage W3 (rows 128..191 of W_edge) swizzled into B-fragment layout
    // fragment (t,n), lane l holds W3[t*4 + 2*(l>>4) + {0,1}][n*16 + (l&15)]
    const float* W3 = W_edge + 2 * NODE_DIM * HIDDEN;
    for (int idx = tid; idx < 16 * 8 * 32; idx += 1024) {
        int t  = idx >> 8;
        int n  = (idx >> 5) & 7;
        int l  = idx & 31;
        int k0 = t * 4 + ((l >> 4) << 1);
        int col = n * 16 + (l & 15);
        lds_w3[idx * 2 + 0] = W3[(size_t)k0 * HIDDEN + col];
        lds_w3[idx * 2 + 1] = W3[(size_t)(k0 + 1) * HIDDEN + col];
    }
    if (tid < HIDDEN) lds_msg[tid] = 0.f;
    __syncthreads();

    const int j0 = wave * 16;
    // A-fragment source: row (b,i,j0+l16) of edge_feature, this lane's K slot
    const float* arow = ef + ((size_t)bi * NN + j0 + l16) * EDGE_DIM + hf * 2;

    const v8f vzero = {0.f, 0.f, 0.f, 0.f, 0.f, 0.f, 0.f, 0.f};
    v8f acc[8];
#pragma unroll
    for (int n = 0; n < 8; ++n) acc[n] = vzero;

    // --- main loop: 16 k-steps x 8 n-tiles = 128 v_wmma_f32_16x16x4_f32
#pragma unroll 4
    for (int t = 0; t < 16; ++t) {
        v2f a = *(const v2f*)(arow + t * 4);
#pragma unroll
        for (int n = 0; n < 8; ++n) {
            v2f bfrag = *(const v2f*)(&lds_w3[(((t << 3) + n) << 6) + lane * 2]);
            acc[n] = __builtin_amdgcn_wmma_f32_16x16x4_f32(
                false, a, false, bfrag, (short)0, acc[n], false, false);
        }
    }

    // --- epilogue: bias adds, ReLU, edge_h store, fused message reduction
    float adjv[8];
    const float* adjrow = adj + (size_t)bi * NN;
#pragma unroll
    for (int r = 0; r < 8; ++r) adjv[r] = adjrow[j0 + r + hf * 8];

    const float* hirow = hi + (size_t)bi * HIDDEN;
    const int b = bi / NN;
    const float* hjb = hj + (size_t)b * NN * HIDDEN;
    float* ehb = edge_h + ((size_t)bi * NN + j0) * HIDDEN;

#pragma unroll
    for (int n = 0; n < 8; ++n) {
        const int col = n * 16 + l16;
        const float hval = hirow[col];
        float msum = 0.f;
#pragma unroll
        for (int r = 0; r < 8; ++r) {
            const int m = r + hf * 8;                 // j offset within tile
            float e = acc[n][r] + hval + hjb[(size_t)(j0 + m) * HIDDEN + col];
            e = fmaxf(e, 0.f);
            ehb[(size_t)m * HIDDEN + col] = e;
            msum = fmaf(adjv[r], e, msum);
        }
        // lanes L and L^16 share the same output column -> combine, then LDS add
        msum += __shfl_xor(msum, 16, 32);
        if (hf == 0) atomicAdd(&lds_msg[col], msum);
    }
    __syncthreads();
    if (tid < HIDDEN) message[(size_t)bi * HIDDEN + tid] = lds_msg[tid];
}

// ---------------------------------------------------------------------------
// Kernel C: ending = relu(message @ Wf[:128] + node_feature @ Wf[128:] + b_feat)
// 1024 rows, K=192, N=128. 16 blocks x 4 waves; wave = 16 rows x 128 cols.
// B-fragments read directly from global (W_feat is 96 KB, L2-resident).
// ---------------------------------------------------------------------------
__global__ __launch_bounds__(128) void ending_kernel(
    const float* __restrict__ nf, const float* __restrict__ message,
    const float* __restrict__ W_feat, const float* __restrict__ b_feat,
    float* __restrict__ ending)
{
    const int tid  = threadIdx.x;
    const int lane = tid & 31;
    const int wave = tid >> 5;          // 0..3
    const int hf   = lane >> 4;
    const int l16  = lane & 15;

    const int row0 = (blockIdx.x * 4 + wave) * 16;
    const int r    = row0 + l16;

    const float* mrow = message + (size_t)r * HIDDEN + hf * 2;
    const float* frow = nf + (size_t)r * NODE_DIM + hf * 2;

    const v8f vzero = {0.f, 0.f, 0.f, 0.f, 0.f, 0.f, 0.f, 0.f};
    v8f acc[8];
#pragma unroll
    for (int n = 0; n < 8; ++n) acc[n] = vzero;

    // K = 0..127 : message part
#pragma unroll 4
    for (int t = 0; t < 32; ++t) {
        v2f a = *(const v2f*)(mrow + t * 4);
        const int k0 = t * 4 + hf * 2;
#pragma unroll
        for (int n = 0; n < 8; ++n) {
            const int col = n * 16 + l16;
            v2f bfrag;
            bfrag[0] = W_feat[(size_t)k0 * HIDDEN + col];
            bfrag[1] = W_feat[(size_t)(k0 + 1) * HIDDEN + col];
            acc[n] = __builtin_amdgcn_wmma_f32_16x16x4_f32(
                false, a, false, bfrag, (short)0, acc[n], false, false);
        }
    }
    // K = 128..191 : node_feature part
#pragma unroll 4
    for (int t = 0; t < 16; ++t) {
        v2f a = *(const v2f*)(frow + t * 4);
        const int k0 = 128 + t * 4 + hf * 2;
#pragma unroll
        for (int n = 0; n < 8; ++n) {
            const int col = n * 16 + l16;
            v2f bfrag;
            bfrag[0] = W_feat[(size_t)k0 * HIDDEN + col];
            bfrag[1] = W_feat[(size_t)(k0 + 1) * HIDDEN + col];
            acc[n] = __builtin_amdgcn_wmma_f32_16x16x4_f32(
                false, a, false, bfrag, (short)0, acc[n], false, false);
        }
    }

#pragma unroll
    for (int n = 0; n < 8; ++n) {
        const int col = n * 16 + l16;
        const float bv = b_feat[col];
#pragma unroll
        for (int rr = 0; rr < 8; ++rr) {
            const int m = row0 + rr + hf * 8;
            ending[(size_t)m * HIDDEN + col] = fmaxf(acc[n][rr] + bv, 0.f);
        }
    }
}

// ---------------------------------------------------------------------------
extern "C" void kernel_launch(void* const* d_in, const int* in_sizes, int n_in,
                              void* d_out, int out_size, void* d_ws, size_t ws_size,
                              hipStream_t stream)
{
    const float* nf  = (const float*)d_in[0];   // node_feature  (B,N,64)
    const float* adj = (const float*)d_in[1];   // adjacency     (B,N,N)
    const float* tgt = (const float*)d_in[2];   // target_info   (B,N,64)
    const float* ef  = (const float*)d_in[3];   // edge_feature  (B,N,N,64)
    // d_in[4] = node_mask (unused by reference math)
    const float* We  = (const float*)d_in[5];   // W_edge (192,128)
    const float* be  = (const float*)d_in[6];   // b_edge (128)
    const float* Wf  = (const float*)d_in[7];   // W_feat (192,128)
    const float* bfv = (const float*)d_in[8];   // b_feat (128)

    float* out     = (float*)d_out;
    float* ending  = out;                                   // B*N*128
    float* edge_h  = out + (size_t)BB * NN * HIDDEN;        // B*N*N*128
    float* tgt_out = edge_h + (size_t)BB * NN * NN * HIDDEN;

    float* hi  = (float*)d_ws;                              // B*N*128
    float* hj  = hi + (size_t)BB * NN * HIDDEN;             // B*N*128
    float* msg = hj + (size_t)BB * NN * HIDDEN;             // B*N*128

    hij_kernel<<<BB * NN, 128, 0, stream>>>(nf, We, be, hi, hj);
    edge_kernel<<<BB * NN, 1024, 0, stream>>>(ef, adj, We, hi, hj, edge_h, msg);
    ending_kernel<<<(BB * NN) / 64, 128, 0, stream>>>(nf, msg, Wf, bfv, ending);
    hipMemcpyAsync(tgt_out, tgt, (size_t)BB * NN * NODE_DIM * sizeof(float),
                   hipMemcpyDeviceToDevice, stream);
}